// GraphProjection_62491774157341
// MI455X (gfx1250) — compile-verified
//
#include <hip/hip_runtime.h>
#include <hip/hip_bf16.h>

typedef __attribute__((ext_vector_type(2))) float v2f;
typedef __attribute__((ext_vector_type(8))) float v8f;

#define BATCH 4
#define NQ    2562
#define NPAD  2576           // 161 tiles of 16
#define OUTC  463
#define KNN   16

// ---------------------------------------------------------------------------
// Prep: queries -> Q4[b][n] = (x, y, z, 1)
// ---------------------------------------------------------------------------
__global__ void prep_q_kernel(const float* __restrict__ in, float* __restrict__ Q4,
                              int N, int Npad) {
  int t = blockIdx.x * blockDim.x + threadIdx.x;
  int b = blockIdx.y;
  if (t >= Npad) return;
  float x = 0.f, y = 0.f, z = 0.f;
  if (t < N) {
    const float* p = in + ((size_t)b * N + t) * 3;
    x = p[0]; y = p[1]; z = p[2];
  }
  float* o = Q4 + (((size_t)b * Npad + t) << 2);
  o[0] = x; o[1] = y; o[2] = z; o[3] = 1.0f;
}

// ---------------------------------------------------------------------------
// Prep: points [B,3,M] -> P4[b][m] = (-2x, -2y, -2z, |p|^2)
// score = A_row . B_col = -2 p.q + |p|^2  ==  d^2 - |q|^2  (rank-equivalent)
// ---------------------------------------------------------------------------
__global__ void prep_p_kernel(const float* __restrict__ pc, float* __restrict__ P4, int M) {
  int t = blockIdx.x * blockDim.x + threadIdx.x;
  int b = blockIdx.y;
  if (t >= M) return;
  const float* base = pc + (size_t)b * 3 * M;
  float x = base[t], y = base[M + t], z = base[2 * M + t];
  float* o = P4 + (((size_t)b * M + t) << 2);
  o[0] = -2.f * x; o[1] = -2.f * y; o[2] = -2.f * z; o[3] = x * x + y * y + z * z;
}

// ---------------------------------------------------------------------------
// Branchless sorted-ascending top-16 insertion (select chain, one guard branch)
// ---------------------------------------------------------------------------
__device__ __forceinline__ void topk_insert(float (&dist)[KNN], int (&idn)[KNN],
                                            float d, int i) {
  if (d >= dist[KNN - 1]) return;
  float cd = d; int ci = i;
#pragma unroll
  for (int j = 0; j < KNN; ++j) {
    const bool  sm = cd < dist[j];
    const float dj = dist[j];
    const int   ij = idn[j];
    dist[j] = sm ? cd : dj;
    idn[j]  = sm ? ci : ij;
    cd      = sm ? dj : cd;
    ci      = sm ? ij : ci;
  }
}

// ---------------------------------------------------------------------------
// KNN via V_WMMA_F32_16X16X4_F32.
// A (SRC0) = 16 candidates x 4  : lane L holds K = 2*(L>=16) .. +1 of row L&15
// B (SRC1) = 4 x 16 queries     : mirrored per-lane pattern
// D = 16x16 f32: lane L holds query n=L&15, candidates m = r + (L>=16 ? 8 : 0)
// One wave handles one 16-query tile; loops all candidate tiles.
// ---------------------------------------------------------------------------
__global__ void knn16_wmma_kernel(const float* __restrict__ Q4,
                                  const float* __restrict__ P4,
                                  int* __restrict__ idxOut,
                                  int M, int N, int Npad) {
  const int lane  = threadIdx.x & 31;
  const int wave  = threadIdx.x >> 5;
  const int qtile = blockIdx.x * (blockDim.x >> 5) + wave;
  const int numQT = Npad >> 4;
  if (qtile >= numQT) return;          // whole wave exits together (EXEC stays full)
  const int b    = blockIdx.y;
  const int r16  = lane & 15;
  const int kofs = (lane >> 4) << 1;   // 0 or 2
  const int mofs = (lane >> 4) << 3;   // 0 or 8

  // B matrix: query tile, constant across candidate loop
  v2f bq = *(const v2f*)(Q4 + (((size_t)b * Npad + qtile * 16 + r16) << 2) + kofs);

  float dist[KNN]; int idn[KNN];
#pragma unroll
  for (int j = 0; j < KNN; ++j) { dist[j] = 3.0e38f; idn[j] = 0; }

  const float* pb = P4 + (((size_t)b * M) << 2) + kofs;

  // software-pipelined A load: fetch next tile before consuming current WMMA
  v2f a = *(const v2f*)(pb + ((size_t)r16 << 2));
  for (int ct = 0; ct < M; ct += 16) {
    const int ctn = (ct + 16 < M) ? (ct + 16) : ct;
    v2f an = *(const v2f*)(pb + ((size_t)(ctn + r16) << 2));
    v8f c = {};
    v8f d = __builtin_amdgcn_wmma_f32_16x16x4_f32(
        /*neg_a=*/false, a, /*neg_b=*/false, bq,
        /*c_mod=*/(short)0, c, /*reuse_a=*/false, /*reuse_b=*/false);

    // tile-level skip: min of the 8 candidates vs current worst
    float m8 = fminf(fminf(fminf(d[0], d[1]), fminf(d[2], d[3])),
                     fminf(fminf(d[4], d[5]), fminf(d[6], d[7])));
    if (m8 < dist[KNN - 1]) {
#pragma unroll
      for (int r = 0; r < 8; ++r)
        topk_insert(dist, idn, d[r], ct + mofs + r);
    }
    a = an;
  }

  // Merge lane pair (L, L^16): snapshot partner's list first, then insert.
  float od[KNN]; int oi[KNN];
#pragma unroll
  for (int j = 0; j < KNN; ++j) {
    od[j] = __shfl_xor(dist[j], 16, 32);
    oi[j] = __shfl_xor(idn[j], 16, 32);
  }
#pragma unroll
  for (int j = 0; j < KNN; ++j) topk_insert(dist, idn, od[j], oi[j]);

  const int q = qtile * 16 + r16;
  if (lane < 16 && q < N) {
    int* op = idxOut + ((size_t)b * Npad + q) * KNN;
#pragma unroll
    for (int j = 0; j < KNN; ++j) op[j] = idn[j];
  }
}

// ---------------------------------------------------------------------------
// Gather + mean: out[b][n][outOfs + c] = mean_k feat[b][c][idx[b][n][k]]
// (coords use the same [B,3,M] layout as a D=3 feature array)
// ---------------------------------------------------------------------------
__global__ void gather_mean_kernel(const float* __restrict__ feat, int D, int M,
                                   const int* __restrict__ idx,
                                   float* __restrict__ out, int outOfs,
                                   int N, int Npad) {
  long t = (long)blockIdx.x * blockDim.x + threadIdx.x;
  int b = blockIdx.y;
  if (t >= (long)N * D) return;
  int n = (int)(t / D);
  int c = (int)(t % D);
  const int*   ip = idx + ((size_t)b * Npad + n) * KNN;
  const float* f  = feat + ((size_t)b * D + c) * M;
  float s = 0.f;
#pragma unroll
  for (int k = 0; k < KNN; ++k) s += f[ip[k]];
  out[((size_t)b * N + n) * OUTC + outOfs + c] = s * (1.0f / KNN);
}

// ---------------------------------------------------------------------------
// Copy inputs into out[..., 0:3]
// ---------------------------------------------------------------------------
__global__ void copy_in_kernel(const float* __restrict__ in, float* __restrict__ out,
                               int N) {
  int t = blockIdx.x * blockDim.x + threadIdx.x;
  int b = blockIdx.y;
  if (t >= N * 3) return;
  int n = t / 3, d = t % 3;
  out[((size_t)b * N + n) * OUTC + d] = in[((size_t)b * N + n) * 3 + d];
}

// ---------------------------------------------------------------------------
extern "C" void kernel_launch(void* const* d_in, const int* in_sizes, int n_in,
                              void* d_out, int out_size, void* d_ws, size_t ws_size,
                              hipStream_t stream) {
  (void)in_sizes; (void)n_in; (void)out_size; (void)ws_size;

  const float* inputs = (const float*)d_in[0];
  const float* pc0c   = (const float*)d_in[1];
  const float* pc1c   = (const float*)d_in[2];
  const float* pc1f   = (const float*)d_in[3];
  const float* pc2c   = (const float*)d_in[4];
  const float* pc2f   = (const float*)d_in[5];
  const float* pc3c   = (const float*)d_in[6];
  const float* pc3f   = (const float*)d_in[7];
  float* out = (float*)d_out;

  const int M0 = 8192, M1 = 8192, M2 = 4096, M3 = 2048;
  const int D1 = 64, D2 = 128, D3 = 256;

  // Workspace carve-up (floats/ints, all 4-byte elements)
  float* Q4  = (float*)d_ws;
  float* P40 = Q4  + (size_t)BATCH * NPAD * 4;
  float* P41 = P40 + (size_t)BATCH * M0 * 4;
  float* P42 = P41 + (size_t)BATCH * M1 * 4;
  float* P43 = P42 + (size_t)BATCH * M2 * 4;
  int*   ix0 = (int*)(P43 + (size_t)BATCH * M3 * 4);
  int*   ix1 = ix0 + (size_t)BATCH * NPAD * KNN;
  int*   ix2 = ix1 + (size_t)BATCH * NPAD * KNN;
  int*   ix3 = ix2 + (size_t)BATCH * NPAD * KNN;

  dim3 blk(256);

  // Prep
  prep_q_kernel<<<dim3((NPAD + 255) / 256, BATCH), blk, 0, stream>>>(inputs, Q4, NQ, NPAD);
  prep_p_kernel<<<dim3(M0 / 256, BATCH), blk, 0, stream>>>(pc0c, P40, M0);
  prep_p_kernel<<<dim3(M1 / 256, BATCH), blk, 0, stream>>>(pc1c, P41, M1);
  prep_p_kernel<<<dim3(M2 / 256, BATCH), blk, 0, stream>>>(pc2c, P42, M2);
  prep_p_kernel<<<dim3(M3 / 256, BATCH), blk, 0, stream>>>(pc3c, P43, M3);

  // KNN: 8 waves/block, one 16-query tile per wave
  const int numQT = NPAD / 16;                 // 161
  dim3 kgrid((numQT + 7) / 8, BATCH);
  knn16_wmma_kernel<<<kgrid, blk, 0, stream>>>(Q4, P40, ix0, M0, NQ, NPAD);
  knn16_wmma_kernel<<<kgrid, blk, 0, stream>>>(Q4, P41, ix1, M1, NQ, NPAD);
  knn16_wmma_kernel<<<kgrid, blk, 0, stream>>>(Q4, P42, ix2, M2, NQ, NPAD);
  knn16_wmma_kernel<<<kgrid, blk, 0, stream>>>(Q4, P43, ix3, M3, NQ, NPAD);

  // Output assembly
  copy_in_kernel<<<dim3((NQ * 3 + 255) / 256, BATCH), blk, 0, stream>>>(inputs, out, NQ);

  auto gdim = [](long n) { return dim3((unsigned)((n + 255) / 256), BATCH); };
  // s0: coords mean -> cols 3..5
  gather_mean_kernel<<<gdim((long)NQ * 3),   blk, 0, stream>>>(pc0c, 3,  M0, ix0, out, 3,   NQ, NPAD);
  // s1: coords -> 6..8, feat -> 9..72
  gather_mean_kernel<<<gdim((long)NQ * 3),   blk, 0, stream>>>(pc1c, 3,  M1, ix1, out, 6,   NQ, NPAD);
  gather_mean_kernel<<<gdim((long)NQ * D1),  blk, 0, stream>>>(pc1f, D1, M1, ix1, out, 9,   NQ, NPAD);
  // s2: coords -> 73..75, feat -> 76..203
  gather_mean_kernel<<<gdim((long)NQ * 3),   blk, 0, stream>>>(pc2c, 3,  M2, ix2, out, 73,  NQ, NPAD);
  gather_mean_kernel<<<gdim((long)NQ * D2),  blk, 0, stream>>>(pc2f, D2, M2, ix2, out, 76,  NQ, NPAD);
  // s3: coords -> 204..206, feat -> 207..462
  gather_mean_kernel<<<gdim((long)NQ * 3),   blk, 0, stream>>>(pc3c, 3,  M3, ix3, out, 204, NQ, NPAD);
  gather_mean_kernel<<<gdim((long)NQ * D3),  blk, 0, stream>>>(pc3f, D3, M3, ix3, out, 207, NQ, NPAD);
}